// ECTLayer_37838661878456
// MI455X (gfx1250) — compile-verified
//
#include <hip/hip_runtime.h>

typedef __attribute__((ext_vector_type(2))) float v2f;
typedef __attribute__((ext_vector_type(8))) float v8f;

#define B_CT   128
#define T_CT   64
#define R_CT   64
#define SLICES 4
#define WAVES  8
#define RADIUS_F 1.1f

// sigmoid(8*(lin - nh)) = 0.5*tanh(4*(lin - nh)) + 0.5
// arg = lin4[r] - nh4, lin4 = 4*lin[r], nh4 = 4*nh
#define LIN4_BASE (4.0f * (-RADIUS_F))
#define LIN4_STEP (4.0f * (2.0f * RADIUS_F / (float)(R_CT - 1)))

__global__ void ect_zero_kernel(float* __restrict__ out, int n) {
    int idx = blockIdx.x * blockDim.x + threadIdx.x;
    if (idx < n) out[idx] = 0.0f;
}

__global__ __launch_bounds__(WAVES * 32) void ect_accum_kernel(
    const float* __restrict__ x,     // [N,3]
    const float* __restrict__ v,     // [3,64]
    const int*   __restrict__ batch, // [N], sorted, values in [0,B)
    float*       __restrict__ out,   // [B,64,64]
    int N) {
    // One shared accumulator tile per workgroup, [R][T] layout:
    // LDS bank == theta -> the 16 lanes of each half hit 16 distinct banks.
    __shared__ float tile[R_CT * T_CT]; // 16 KB

    const int tid  = threadIdx.x;
    const int lane = tid & 31;
    const int w    = tid >> 5;
    const int i    = lane & 15;  // column index within 16-wide half
    const int h    = lane >> 4;  // lane half

    const int bid = blockIdx.x;
    const int b   = bid / SLICES;
    const int s   = bid % SLICES;

    // Cooperative zero of the shared tile
    for (int k = tid; k < R_CT * T_CT; k += WAVES * 32) tile[k] = 0.0f;

    // Bucket range [n0,n1) via binary search on the sorted batch array (uniform per thread)
    int lo = 0, hi = N;
    while (lo < hi) { int mid = (lo + hi) >> 1; if (batch[mid] < b) lo = mid + 1; else hi = mid; }
    const int n0 = lo;
    hi = N;
    while (lo < hi) { int mid = (lo + hi) >> 1; if (batch[mid] < b + 1) lo = mid + 1; else hi = mid; }
    const int n1 = lo;

    const int ntiles = (n1 - n0 + 15) >> 4;
    const int ts0 = (ntiles * s) / SLICES;
    const int ts1 = (ntiles * (s + 1)) / SLICES;

    // B fragments: 4x16 f32 per theta-block. B[k][t]:
    // lanes 0-15: vgpr0=K0,vgpr1=K1 ; lanes 16-31: vgpr0=K2,vgpr1=K3(pad=0)
    v2f bf[4];
#pragma unroll
    for (int tb = 0; tb < 4; ++tb) {
        int t = tb * 16 + i;
        if (h == 0) { bf[tb].x = v[0 * T_CT + t]; bf[tb].y = v[1 * T_CT + t]; }
        else        { bf[tb].x = v[2 * T_CT + t]; bf[tb].y = 0.0f; }
    }

    __syncthreads(); // tile zeroed before any wave accumulates

    for (int tix = ts0 + w; tix < ts1; tix += WAVES) {
        const int nb = n0 + (tix << 4); // base node of this 16-node tile

        // A fragment: 16x4 f32. lanes 0-15: vgpr0=A[m][0],vgpr1=A[m][1];
        // lanes 16-31: vgpr0=A[m][2], vgpr1=A[m][3]=0 (D padded 3->4).
        // Invalid (tail) nodes are neutralized AFTER the WMMA via nh4=+huge,
        // so we can clamp the index and load unconditionally (no exec masking).
        const int node  = nb + i;
        const int nodec = (node < n1) ? node : (n1 - 1); // loop body implies n1 > nb >= 0
        __builtin_prefetch(&x[(nb + 16) * 3], 0, 1);

        const float e0 = x[nodec * 3 + (h << 1)]; // h=0: x0, h=1: x2
        const float e1 = x[nodec * 3 + 1];        // only used by h=0
        v2f af;
        af.x = e0;
        af.y = h ? 0.0f : e1;

        v8f czero = {};
#pragma unroll
        for (int tb = 0; tb < 4; ++tb) {
            // nh tile: D[m][t] for m=0..15, t=tb*16..tb*16+15 (full fp32)
            v8f acc = __builtin_amdgcn_wmma_f32_16x16x4_f32(
                false, af, false, bf[tb], (short)0, czero, false, false);

            // Per lane: vgpr g holds node m = g + 8*h at theta t = tb*16 + i.
            // Scale by 4 for the tanh form; invalid nodes -> +huge so tanh(-huge) = -1
            // contributes exactly 0 after the 0.5*z+0.5 affine map.
            float nh4[8];
#pragma unroll
            for (int g = 0; g < 8; ++g) {
                int m = g + 8 * h;
                nh4[g] = (nb + m < n1) ? acc[g] * 4.0f : 3.0e38f;
            }

            const int t = tb * 16 + i;
#pragma unroll 4
            for (int r = 0; r < R_CT; ++r) {
                float lin4 = fmaf((float)r, LIN4_STEP, LIN4_BASE);
                float th = __builtin_amdgcn_tanhf(lin4 - nh4[0]);
#pragma unroll
                for (int g = 1; g < 8; ++g)
                    th += __builtin_amdgcn_tanhf(lin4 - nh4[g]);
                // Each half contributes 8 nodes: sum(sigmoid) = 0.5*sum(tanh) + 8*0.5.
                // No-return LDS atomic: no read-back dependency, both halves active;
                // the two halves' atomics merge the 16 nodes of the tile.
                atomicAdd(&tile[r * T_CT + t], fmaf(0.5f, th, 4.0f));
            }
        }
    }

    __syncthreads();

    // Flush: one global f32 atomic per output element per workgroup
    const long long outBase = (long long)b << 12; // b * 64*64
    for (int k = tid; k < R_CT * T_CT; k += WAVES * 32) {
        int t = k & (T_CT - 1);
        int r = k >> 6;
        atomicAdd(&out[outBase + (long long)t * R_CT + r], tile[k]);
    }
}

extern "C" void kernel_launch(void* const* d_in, const int* in_sizes, int n_in,
                              void* d_out, int out_size, void* d_ws, size_t ws_size,
                              hipStream_t stream) {
    const float* x     = (const float*)d_in[0];
    const float* v     = (const float*)d_in[1];
    const int*   batch = (const int*)d_in[2];
    float*       out   = (float*)d_out;
    const int N = in_sizes[2];

    ect_zero_kernel<<<(out_size + 255) / 256, 256, 0, stream>>>(out, out_size);
    ect_accum_kernel<<<B_CT * SLICES, WAVES * 32, 0, stream>>>(x, v, batch, out, N);
}